// BlockLayer_75376676045426
// MI455X (gfx1250) — compile-verified
//
#include <hip/hip_runtime.h>
#include <hip/hip_bf16.h>

// ---------------------------------------------------------------------------
// Graph wavelet GNN layer for MI455X (gfx1250, wave32, WMMA).
//
// Math: w1 = expm(-2L), w2 = expm(-4*w1)  (no eigh needed).
// L spectrum in [0,2] => scale by 1/16, degree-12 Taylor (Horner), square 4x.
// ~30 f32 GEMMs of 2048^3 => compute bound (50MB/GEMM fits in 192MB L2).
// f32 WMMA 16x16x4 keeps the chained-exponential numerics faithful.
// Data movement: double-buffered GLOBAL_LOAD_ASYNC_TO_LDS_B128 (ASYNCcnt)
// when the toolchain exposes the builtin; synchronous b128 staging otherwise.
// ---------------------------------------------------------------------------

typedef float v2f __attribute__((ext_vector_type(2)));
typedef float v8f __attribute__((ext_vector_type(8)));
typedef int   v4i_ldst __attribute__((vector_size(16)));   // builtin's pointee type

#define GN    2048
#define GIN   1024
#define GOUT  1024

#if defined(__has_builtin)
#  if __has_builtin(__builtin_amdgcn_global_load_async_to_lds_b128)
#    define HAVE_ASYNC_LDS 1
#  endif
#endif

#if defined(HAVE_ASYNC_LDS)
__device__ __forceinline__ void async_copy16(const float* g, float* l)
{
    __builtin_amdgcn_global_load_async_to_lds_b128((v4i_ldst*)g, (v4i_ldst*)l, 0, 0);
}
__device__ __forceinline__ void async_wait3()
{
#  if __has_builtin(__builtin_amdgcn_s_wait_asynccnt)
    __builtin_amdgcn_s_wait_asynccnt(3);
#  else
    asm volatile("s_wait_asynccnt 0x3" ::: "memory");
#  endif
}
__device__ __forceinline__ void async_wait0()
{
#  if __has_builtin(__builtin_amdgcn_s_wait_asynccnt)
    __builtin_amdgcn_s_wait_asynccnt(0);
#  else
    asm volatile("s_wait_asynccnt 0x0" ::: "memory");
#  endif
}
#else
__device__ __forceinline__ void async_copy16(const float* g, float* l)
{
    *(float4*)l = *(const float4*)g;   // sync fallback: b128 load + b128 ds store
}
__device__ __forceinline__ void async_wait3() {}
__device__ __forceinline__ void async_wait0() {}
#endif

// ------------------------------ WMMA GEMM ----------------------------------
// C[M,N] = alpha*A[M,K]@B[K,N] + beta*C + diag*I + bias  (row-major, NN)
// Requires M%128==0, N%64==0, K%16==0 (true for all uses here).
// Block: 256 threads = 8 waves; tile 128x64; wave grid 4x2, 32x32 per wave.
// Fragment layouts per CDNA5 ISA 7.12.2:
//   A 16x4 f32 : lane l, vgpr j -> A[l%16][2*(l/16)+j]   (one ds_load_b64)
//   B 4x16 f32 : lane l, vgpr v -> B[2*(l/16)+v][l%16]
//   C 16x16 f32: lane l, vgpr v -> C[v+8*(l/16)][l%16]
// LDS: natural layouts, strides chosen for b128-aligned async stores and
// conflict-free b64/b32 fragment reads (A stride 20, B stride 80 floats).
__launch_bounds__(256)
__global__ void gemm_wmma_f32(float* __restrict__ C, const float* __restrict__ A,
                              const float* __restrict__ B, int M, int Nn, int K,
                              float alpha, float beta, float diag,
                              const float* __restrict__ bias)
{
    constexpr int BM = 128, BN = 64, KT = 16;
    constexpr int LDA = 20;   // floats; 80B row stride (16B multiple, bank-clean)
    constexpr int LDB = 80;   // floats; 320B row stride (lane halves 32 banks apart)
    __shared__ __align__(16) float As[2][BM * LDA];
    __shared__ __align__(16) float Bs[2][KT * LDB];

    const int tid  = threadIdx.x;
    const int lane = tid & 31;
    const int wave = tid >> 5;
    const int wm   = wave & 3;            // row group (0..3) -> 32 rows each
    const int wn   = wave >> 2;           // col group (0..1) -> 32 cols each
    const int row0 = blockIdx.y * BM;
    const int col0 = blockIdx.x * BN;
    const int l16  = lane & 15;
    const int lh   = lane >> 4;

    // staging: A tile 128x16 = 512 x 16B chunks (2/thread), B tile 16x64 = 256 (1/thread)
    const int rA = tid >> 2, cA = (tid & 3) << 2;     // + second chunk at rA+64
    const int rB = tid >> 4, cB = (tid & 15) << 2;

    auto stage = [&](int buf, int kk0) {
        async_copy16(A + (size_t)(row0 + rA) * K + kk0 + cA,      &As[buf][rA * LDA + cA]);
        async_copy16(A + (size_t)(row0 + rA + 64) * K + kk0 + cA, &As[buf][(rA + 64) * LDA + cA]);
        async_copy16(B + (size_t)(kk0 + rB) * Nn + col0 + cB,     &Bs[buf][rB * LDB + cB]);
    };

    v8f acc[2][2];
    #pragma unroll
    for (int i = 0; i < 2; ++i)
        #pragma unroll
        for (int j = 0; j < 2; ++j)
            #pragma unroll
            for (int v = 0; v < 8; ++v) acc[i][j][v] = 0.0f;

    const int nt = K / KT;
    stage(0, 0);                                   // preload tile 0

    for (int t = 0; t < nt; ++t) {
        const int buf = t & 1;
        if (t + 1 < nt) {
            stage(buf ^ 1, (t + 1) * KT);          // prefetch next tile (async)
            async_wait3();                         // tile t's 3 asyncs retired (in-order)
        } else {
            async_wait0();
        }
        __syncthreads();                           // all waves' tile-t data visible

        const float* Ab = As[buf];
        const float* Bb = Bs[buf];
        #pragma unroll
        for (int kk = 0; kk < KT; kk += 4) {
            const int kb = kk + 2 * lh;            // K pair base for this lane half
            v2f a0 = *(const v2f*)(Ab + (32 * wm + l16) * LDA + kb);
            v2f a1 = *(const v2f*)(Ab + (32 * wm + 16 + l16) * LDA + kb);
            v2f b0, b1;
            b0.x = Bb[(kb + 0) * LDB + 32 * wn + l16];
            b0.y = Bb[(kb + 1) * LDB + 32 * wn + l16];
            b1.x = Bb[(kb + 0) * LDB + 32 * wn + 16 + l16];
            b1.y = Bb[(kb + 1) * LDB + 32 * wn + 16 + l16];
            acc[0][0] = __builtin_amdgcn_wmma_f32_16x16x4_f32(false, a0, false, b0, (short)0, acc[0][0], false, false);
            acc[0][1] = __builtin_amdgcn_wmma_f32_16x16x4_f32(false, a0, false, b1, (short)0, acc[0][1], false, false);
            acc[1][0] = __builtin_amdgcn_wmma_f32_16x16x4_f32(false, a1, false, b0, (short)0, acc[1][0], false, false);
            acc[1][1] = __builtin_amdgcn_wmma_f32_16x16x4_f32(false, a1, false, b1, (short)0, acc[1][1], false, false);
        }
        __syncthreads();                           // done reading buf before overwrite
    }

    // ---- epilogue: C = alpha*acc + beta*C + diag*I + bias ----------------
    #pragma unroll
    for (int i = 0; i < 2; ++i) {
        #pragma unroll
        for (int j = 0; j < 2; ++j) {
            #pragma unroll
            for (int v = 0; v < 8; ++v) {
                const int gr = row0 + 32 * wm + 16 * i + 8 * lh + v;
                const int gc = col0 + 32 * wn + 16 * j + l16;
                const size_t idx = (size_t)gr * Nn + gc;
                float val = alpha * acc[i][j][v];
                if (beta != 0.0f) val += beta * C[idx];
                if (gr == gc)     val += diag;
                if (bias)         val += bias[gc];
                C[idx] = val;
            }
        }
    }
}

// --------------------------- small kernels ---------------------------------

__global__ void k_rowsum_dinv(const float* __restrict__ adj, float* __restrict__ dinv, int n)
{
    __shared__ float red[256];
    const int i = blockIdx.x;
    float s = 0.f;
    for (int j = threadIdx.x; j < n; j += 256) s += adj[(size_t)i * n + j];
    red[threadIdx.x] = s; __syncthreads();
    for (int off = 128; off > 0; off >>= 1) {
        if (threadIdx.x < off) red[threadIdx.x] += red[threadIdx.x + off];
        __syncthreads();
    }
    if (threadIdx.x == 0) dinv[i] = rsqrtf(red[0]);
}

// X = scale * (I - dinv_i * adj_ij * dinv_j)
__global__ void k_build_X1(float* __restrict__ X, const float* __restrict__ adj,
                           const float* __restrict__ dinv, int n, float scale)
{
    const size_t idx = (size_t)blockIdx.x * 256 + threadIdx.x;
    const int i = (int)(idx / n), j = (int)(idx % n);
    const float lij = (i == j ? 1.f : 0.f) - dinv[i] * adj[idx] * dinv[j];
    X[idx] = scale * lij;
}

// dst = scale*src + diag*I
__global__ void k_axpdiag(float* __restrict__ dst, const float* __restrict__ src,
                          int n, float scale, float diag)
{
    const size_t idx = (size_t)blockIdx.x * 256 + threadIdx.x;
    const int i = (int)(idx / n), j = (int)(idx % n);
    dst[idx] = scale * src[idx] + (i == j ? diag : 0.f);
}

__global__ void k_zero(float* __restrict__ p)
{
    p[(size_t)blockIdx.x * 256 + threadIdx.x] = 0.f;
}

__global__ void k_init_nodes(unsigned* __restrict__ m_enc, float* __restrict__ denom, int n)
{
    const int i = blockIdx.x * 256 + threadIdx.x;
    if (i < n) { m_enc[i] = 0x007FFFFFu; /* enc(-inf) */ denom[i] = 0.f; }
}

// per-node dot products: a1[i] = z[i].aw[0:1024], a2[i] = z[i].aw[1024:2048]
__global__ void k_node_attn(const float* __restrict__ z, const float* __restrict__ aw,
                            float* __restrict__ a1, float* __restrict__ a2)
{
    __shared__ float r1[256], r2[256];
    const int i = blockIdx.x;
    float s1 = 0.f, s2 = 0.f;
    for (int c = threadIdx.x; c < GOUT; c += 256) {
        const float zv = z[(size_t)i * GOUT + c];
        s1 += zv * aw[c];
        s2 += zv * aw[GOUT + c];
    }
    r1[threadIdx.x] = s1; r2[threadIdx.x] = s2; __syncthreads();
    for (int off = 128; off > 0; off >>= 1) {
        if (threadIdx.x < off) {
            r1[threadIdx.x] += r1[threadIdx.x + off];
            r2[threadIdx.x] += r2[threadIdx.x + off];
        }
        __syncthreads();
    }
    if (threadIdx.x == 0) { a1[i] = r1[0]; a2[i] = r2[0]; }
}

__device__ __forceinline__ unsigned f32_enc(float f)
{
    const unsigned b = __float_as_uint(f);
    return (b & 0x80000000u) ? ~b : (b | 0x80000000u);
}
__device__ __forceinline__ float f32_dec(unsigned u)
{
    const unsigned b = (u & 0x80000000u) ? (u ^ 0x80000000u) : ~u;
    return __uint_as_float(b);
}

__global__ void k_edges1(const float* __restrict__ e, const int* __restrict__ src,
                         const int* __restrict__ dst, const float* __restrict__ a1,
                         const float* __restrict__ a2, const float* __restrict__ ew,
                         const float* __restrict__ aw_tail, float* __restrict__ logits,
                         float* __restrict__ ex, unsigned* __restrict__ m_enc, int E)
{
    const int eidx = blockIdx.x * 256 + threadIdx.x;
    if (eidx >= E) return;
    const float e0 = e[2 * eidx], e1 = e[2 * eidx + 1];
    const float x0 = e0 * ew[0] + e1 * ew[1];      // ex = e @ edge_w.T
    const float x1 = e0 * ew[2] + e1 * ew[3];
    float lg = a1[src[eidx]] + a2[dst[eidx]] + x0 * aw_tail[0] + x1 * aw_tail[1];
    lg = lg > 0.f ? lg : 0.01f * lg;               // leaky_relu
    logits[eidx] = lg;
    ex[2 * eidx] = x0; ex[2 * eidx + 1] = x1;
    atomicMax(&m_enc[dst[eidx]], f32_enc(lg));     // segment max (order-preserving enc)
}

__global__ void k_edges2(const float* __restrict__ logits, const int* __restrict__ dst,
                         const unsigned* __restrict__ m_enc, float* __restrict__ p,
                         float* __restrict__ denom, int E)
{
    const int eidx = blockIdx.x * 256 + threadIdx.x;
    if (eidx >= E) return;
    const int d = dst[eidx];
    const float pe = __expf(logits[eidx] - f32_dec(m_enc[d]));
    p[eidx] = pe;
    atomicAdd(&denom[d], pe);
}

// out[dst] += (p/denom[dst]) * (z[src] + ex @ e2n_w.T); one block per edge
__global__ void k_scatter(const float* __restrict__ z, const int* __restrict__ src,
                          const int* __restrict__ dst, const float* __restrict__ p,
                          const float* __restrict__ denom, const float* __restrict__ ex,
                          const float* __restrict__ e2n, float* __restrict__ out)
{
    const int eidx = blockIdx.x;
    const int d = dst[eidx], s = src[eidx];
    const float coef = p[eidx] / denom[d];
    const float x0 = ex[2 * eidx], x1 = ex[2 * eidx + 1];
    const int c0 = threadIdx.x * 4;
    const float4 zv  = *(const float4*)(z + (size_t)s * GOUT + c0);
    const float4 w01 = *(const float4*)(e2n + 2 * c0);      // rows c0,c0+1 of [1024][2]
    const float4 w23 = *(const float4*)(e2n + 2 * c0 + 4);  // rows c0+2,c0+3
    float* o = out + (size_t)d * GOUT + c0;
    atomicAdd(o + 0, coef * (zv.x + x0 * w01.x + x1 * w01.y));
    atomicAdd(o + 1, coef * (zv.y + x0 * w01.z + x1 * w01.w));
    atomicAdd(o + 2, coef * (zv.z + x0 * w23.x + x1 * w23.y));
    atomicAdd(o + 3, coef * (zv.w + x0 * w23.z + x1 * w23.w));
}

// ------------------------------ launch -------------------------------------

extern "C" void kernel_launch(void* const* d_in, const int* in_sizes, int n_in,
                              void* d_out, int out_size, void* d_ws, size_t ws_size,
                              hipStream_t stream)
{
    const float* h    = (const float*)d_in[0];
    const float* e    = (const float*)d_in[1];
    const float* adj  = (const float*)d_in[2];
    const int*   src  = (const int*)d_in[3];
    const int*   dst  = (const int*)d_in[4];
    const float* W1w  = (const float*)d_in[5];
    const float* W2w  = (const float*)d_in[6];
    const float* W3w  = (const float*)d_in[7];
    const float* bias = (const float*)d_in[8];
    const float* aw   = (const float*)d_in[9];
    const float* ew   = (const float*)d_in[10];
    const float* e2n  = (const float*)d_in[11];
    float* out = (float*)d_out;
    const int E = in_sizes[3];
    const int n = GN;
    const size_t NN = (size_t)n * n;

    // workspace layout (f32)
    float* ws    = (float*)d_ws;
    float* Xm    = ws;              // scaled matrix for expm
    float* Ta    = Xm  + NN;
    float* Tb    = Ta  + NN;
    float* W1m   = Tb  + NN;        // expm(-2L)
    float* w1h   = W1m + NN;        // [N, GIN]
    float* w2h   = w1h + (size_t)n * GIN;
    float* z     = w2h + (size_t)n * GIN;  // [N, GOUT]
    float* dinv  = z   + (size_t)n * GOUT;
    float* a1    = dinv + n;
    float* a2    = a1 + n;
    float* denom = a2 + n;
    unsigned* m_enc = (unsigned*)(denom + n);
    float* logits   = (float*)(m_enc + n);
    float* pbuf     = logits + E;
    float* exbuf    = pbuf + E;

    auto gemm = [&](float* C, const float* A, const float* B, int M, int Nc, int K,
                    float alpha, float beta, float diag, const float* b_) {
        dim3 g(Nc / 64, M / 128);
        gemm_wmma_f32<<<g, 256, 0, stream>>>(C, A, B, M, Nc, K, alpha, beta, diag, b_);
    };

    // expm of Xm (||Xm|| <= 0.25): degree-12 Taylor (Horner) + 4 squarings.
    auto expm_scaled = [&]() -> float* {
        const int nblk = (int)(NN / 256);
        k_axpdiag<<<nblk, 256, 0, stream>>>(Ta, Xm, n, 1.f / 12.f, 1.f);  // T = I + X/12
        float* cur = Ta; float* nxt = Tb;
        for (int k = 11; k >= 1; --k) {            // T = (X@T)/k + I
            gemm(nxt, Xm, cur, n, n, n, 1.f / (float)k, 0.f, 1.f, nullptr);
            float* t = cur; cur = nxt; nxt = t;
        }
        for (int i = 0; i < 4; ++i) {              // T = T@T  (x4 => ^16)
            gemm(nxt, cur, cur, n, n, n, 1.f, 0.f, 0.f, nullptr);
            float* t = cur; cur = nxt; nxt = t;
        }
        return cur;
    };

    // 1) normalized Laplacian, scaled: Xm = -2L/16 = -0.125*(I - D^-1/2 A D^-1/2)
    k_rowsum_dinv<<<n, 256, 0, stream>>>(adj, dinv, n);
    k_build_X1<<<(int)(NN / 256), 256, 0, stream>>>(Xm, adj, dinv, n, -0.125f);

    // 2) w1 = expm(-2L)
    float* r1 = expm_scaled();
    (void)hipMemcpyAsync(W1m, r1, NN * sizeof(float), hipMemcpyDeviceToDevice, stream);

    // 3) w2 = expm(-4*w1): Xm = -w1/4
    k_axpdiag<<<(int)(NN / 256), 256, 0, stream>>>(Xm, W1m, n, -0.25f, 0.f);
    float* W2m = expm_scaled();

    // 4) wavelet-filtered features
    gemm(w1h, W1m, h, n, GIN, n, 1.f, 0.f, 0.f, nullptr);
    gemm(w2h, W2m, h, n, GIN, n, 1.f, 0.f, 0.f, nullptr);

    // 5) z = h@W + w1h@W2 + w2h@W3 + bias
    gemm(z, h,   W1w, n, GOUT, GIN,  1.f, 0.f, 0.f, nullptr);
    gemm(z, w1h, W2w, n, GOUT, GOUT, 1.f, 1.f, 0.f, nullptr);
    gemm(z, w2h, W3w, n, GOUT, GOUT, 1.f, 1.f, 0.f, bias);

    // 6) edge attention + segment softmax + scatter
    k_node_attn<<<n, 256, 0, stream>>>(z, aw, a1, a2);
    k_init_nodes<<<(n + 255) / 256, 256, 0, stream>>>(m_enc, denom, n);
    k_zero<<<(int)((size_t)n * GOUT / 256), 256, 0, stream>>>(out);
    k_edges1<<<(E + 255) / 256, 256, 0, stream>>>(e, src, dst, a1, a2, ew, aw + 2 * GOUT,
                                                  logits, exbuf, m_enc, E);
    k_edges2<<<(E + 255) / 256, 256, 0, stream>>>(logits, dst, m_enc, pbuf, denom, E);
    k_scatter<<<E, 256, 0, stream>>>(z, src, dst, pbuf, denom, exbuf, e2n, out);
}